// ScaleDotProductAttention_4294967296400
// MI455X (gfx1250) — compile-verified
//
#include <hip/hip_runtime.h>

typedef __attribute__((ext_vector_type(2))) float v2f;
typedef __attribute__((ext_vector_type(8))) float v8f;

// Problem constants (match reference: B=2, H=16, S=2048, D=64, fp32).
constexpr int cB = 2;
constexpr int cH = 16;
constexpr int cS = 2048;
constexpr int cD = 64;
constexpr int QT = 16;              // query rows per workgroup
constexpr int NKT = cS / 16;        // 128 key tiles of 16
constexpr float SCALE = 0.125f;     // 1/sqrt(64)

// One workgroup = 256 threads = 8 wave32. Handles one 16-row query tile of one (b,h).
__global__ __launch_bounds__(256, 2)
void ScaleDotProductAttention_4294967296400_kernel(
    const float* __restrict__ Q, const float* __restrict__ K,
    const float* __restrict__ V, float* __restrict__ Out,
    float* __restrict__ Attn)
{
    // LDS: full 16 x 2048 fp32 score strip (128 KB) + reduction scratch + split-K partials.
    __shared__ float sc[QT * cS];          // 131072 B
    __shared__ float red[16 * 16];         // 1 KB
    __shared__ float rowmax[16];
    __shared__ float rowsum[16];
    __shared__ float part[4 * 256];        // 4 KB  (4 out-tiles of 16x16 partials)

    const int tid  = threadIdx.x;
    const int lane = tid & 31;             // wave32
    const int wave = tid >> 5;             // 0..7
    const int bh   = blockIdx.y;           // 0..31
    const int q0   = blockIdx.x * QT;      // query row base within (b,h)

    const float* Qb = Q + (size_t)bh * cS * cD;
    const float* Kb = K + (size_t)bh * cS * cD;
    const float* Vb = V + (size_t)bh * cS * cD;

    const int mrow = lane & 15;            // matrix row / B-col held by this lane
    const int g    = lane >> 4;            // half-wave group (selects K pair)

    // ---- Q tile as 16 A-fragments (16x4 f32 slices). A layout:
    //   lane L (L<16): row M=L, elems K = 4*s + {0,1}; lane L>=16: K = 4*s + {2,3}.
    v2f aq[16];
    {
        const float* qrow = Qb + (size_t)(q0 + mrow) * cD;
        #pragma unroll
        for (int s = 0; s < 16; ++s) {
            const int kb = s * 4 + g * 2;
            aq[s].x = qrow[kb];
            aq[s].y = qrow[kb + 1];
        }
    }

    // ---- Phase 1: scores = (Q K^T) * scale  ->  LDS strip.
    // Waves stripe over the 128 key tiles; 16 chained f32 WMMAs reduce D=64.
    for (int kt = wave; kt < NKT; kt += 8) {
        const float* krow = Kb + (size_t)(kt * 16 + mrow) * cD;  // B col = key row
        v8f c = {};
        #pragma unroll
        for (int s = 0; s < 16; ++s) {
            const int kb = s * 4 + g * 2;
            v2f b;
            b.x = krow[kb];
            b.y = krow[kb + 1];
            c = __builtin_amdgcn_wmma_f32_16x16x4_f32(
                    false, aq[s], false, b, (short)0, c, false, false);
        }
        // C/D layout: VGPR r -> rows {r, r+8}; lanes 0-15 N=lane, 16-31 N=lane-16.
        #pragma unroll
        for (int r = 0; r < 8; ++r)
            sc[(r + 8 * g) * cS + kt * 16 + mrow] = c[r] * SCALE;
    }
    __syncthreads();

    // ---- Phase 2: row softmax in LDS; write normalized attn to global.
    {
        const int row = tid >> 4;          // 0..15
        const int sub = tid & 15;          // 16 threads cooperate per row

        float m = -INFINITY;
        for (int ccol = sub; ccol < cS; ccol += 16)
            m = fmaxf(m, sc[row * cS + ccol]);
        red[row * 16 + sub] = m;
        __syncthreads();
        if (sub == 0) {
            float mm = red[row * 16];
            #pragma unroll
            for (int i = 1; i < 16; ++i) mm = fmaxf(mm, red[row * 16 + i]);
            rowmax[row] = mm;
        }
        __syncthreads();

        const float rm = rowmax[row];
        float ssum = 0.0f;
        for (int ccol = sub; ccol < cS; ccol += 16) {
            const float e = __expf(sc[row * cS + ccol] - rm);
            sc[row * cS + ccol] = e;
            ssum += e;
        }
        red[row * 16 + sub] = ssum;
        __syncthreads();
        if (sub == 0) {
            float s2 = 0.0f;
            #pragma unroll
            for (int i = 0; i < 16; ++i) s2 += red[row * 16 + i];
            rowsum[row] = s2;
        }
        __syncthreads();

        const float inv = 1.0f / rowsum[row];
        float* attn_row = Attn + ((size_t)bh * cS + q0 + row) * cS;
        for (int ccol = sub; ccol < cS; ccol += 16) {
            const float p = sc[row * cS + ccol] * inv;
            sc[row * cS + ccol] = p;
            attn_row[ccol] = p;              // coalesced: 16 consecutive floats/group
        }
    }
    __syncthreads();

    // ---- Phase 3: Out = attn @ V. 4 out-tiles of 16 cols; split-K across wave pairs.
    {
        const int nt = wave & 3;            // output column tile
        const int kh = wave >> 2;           // K half: 0 -> [0,1024), 1 -> [1024,2048)
        const int n0 = nt * 16;
        const int kbeg = kh * (cS / 2);

        v8f c = {};
        for (int k4 = 0; k4 < cS / 2; k4 += 4) {
            const int kb = kbeg + k4 + g * 2;
            v2f a;                           // A from LDS probability strip
            a.x = sc[mrow * cS + kb];
            a.y = sc[mrow * cS + kb + 1];
            v2f b;                           // B = V slice (4x16), rows strided by D
            const float* vrow = Vb + (size_t)kb * cD + n0 + mrow;
            b.x = vrow[0];
            b.y = vrow[cD];
            c = __builtin_amdgcn_wmma_f32_16x16x4_f32(
                    false, a, false, b, (short)0, c, false, false);
        }

        if (kh == 1) {
            #pragma unroll
            for (int r = 0; r < 8; ++r)
                part[nt * 256 + (r + 8 * g) * 16 + mrow] = c[r];
        }
        __syncthreads();
        if (kh == 0) {
            float* orow = Out + ((size_t)bh * cS + q0) * cD + n0;
            #pragma unroll
            for (int r = 0; r < 8; ++r) {
                const int mr = r + 8 * g;
                orow[(size_t)mr * cD + mrow] =
                    c[r] + part[nt * 256 + mr * 16 + mrow];
            }
        }
    }
}

extern "C" void kernel_launch(void* const* d_in, const int* in_sizes, int n_in,
                              void* d_out, int out_size, void* d_ws, size_t ws_size,
                              hipStream_t stream) {
    const float* q = (const float*)d_in[0];
    const float* k = (const float*)d_in[1];
    const float* v = (const float*)d_in[2];

    float* out  = (float*)d_out;                                   // [B,H,S,D]
    float* attn = (float*)d_out + (size_t)cB * cH * cS * cD;       // [B,H,S,S]

    dim3 grid(cS / QT, cB * cH);   // (128, 32)
    dim3 block(256);
    ScaleDotProductAttention_4294967296400_kernel<<<grid, block, 0, stream>>>(
        q, k, v, out, attn);
}